// GCN_25340307046434
// MI455X (gfx1250) — compile-verified
//
#include <hip/hip_runtime.h>
#include <hip/hip_bf16.h>
#include <math.h>

typedef __attribute__((ext_vector_type(2))) float v2f;
typedef __attribute__((ext_vector_type(8))) float v8f;

// ---------------------------------------------------------------------------
// Degree: deg[dst[e]] += 1  (self-loop +1 folded into dinv kernel)
// ---------------------------------------------------------------------------
__global__ void gcn_degree(const long long* __restrict__ dst, float* __restrict__ deg,
                           long long E) {
    long long i = (long long)blockIdx.x * blockDim.x + threadIdx.x;
    long long stride = (long long)gridDim.x * blockDim.x;
    for (; i < E; i += stride) {
        atomicAdd(&deg[(int)dst[i]], 1.0f);
    }
}

__global__ void gcn_dinv(float* __restrict__ deg, int N) {
    int i = blockIdx.x * blockDim.x + threadIdx.x;
    int stride = gridDim.x * blockDim.x;
    for (; i < N; i += stride) {
        deg[i] = rsqrtf(deg[i] + 1.0f);   // deg>0 always (self-loop)
    }
}

// ---------------------------------------------------------------------------
// WMMA f32 GEMM: D[M x Npad] = A[M x Kpad] * B[Kreal x Nreal], Npad = 16*TN.
// B is staged to LDS in *fragment order*: ldsB[kp*Npad + n] = (B[2kp][n],
// B[2kp+1][n]) as float2, zero padded. The per-step B fragment is then one
// aligned ds_load_b64 directly into the VGPR pair WMMA consumes (no repacks).
// One wave computes a full 16 x Npad strip (TN accumulator tiles); the A
// fragment (global_load_b64) is software-pipelined one K-step ahead.
//
// A frag (16x4 f32): lane<16: row M=lane,    K=k0+0 / k0+1 in the 2 VGPRs
//                    lane>=16: row M=lane-16, K=k0+2 / k0+3
// B frag (4x16 f32): VGPR0 = rows k0+0 (lanes 0-15) / k0+2 (lanes 16-31),
//                    VGPR1 = rows k0+1 / k0+3, column N = n0 + (lane&15)
// C/D (16x16 f32):   VGPR v -> row m0 + v + 8*(lane>>4), col n0 + (lane&15)
// ---------------------------------------------------------------------------
template <int TN>
__global__ void gcn_wmma_gemm(const float* __restrict__ A, int lda,
                              const float* __restrict__ B, int ldb,
                              int Nreal, int Kreal, int Kpad,
                              float* __restrict__ D, int ldd,
                              int M) {
    extern __shared__ v2f ldsB[];                   // [Kpad/2][Npad] float2 pairs
    const int Npad = TN * 16;

    int pairs = (Kpad >> 1) * Npad;
    for (int idx = threadIdx.x; idx < pairs; idx += blockDim.x) {
        int kp = idx / Npad;
        int n = idx - kp * Npad;
        int k = kp << 1;
        v2f p;
        p.x = (k     < Kreal && n < Nreal) ? B[(size_t)k * ldb + n]       : 0.f;
        p.y = (k + 1 < Kreal && n < Nreal) ? B[(size_t)(k + 1) * ldb + n] : 0.f;
        ldsB[idx] = p;
    }
    __syncthreads();

    int wavesPerBlock = blockDim.x >> 5;
    int strip = blockIdx.x * wavesPerBlock + (threadIdx.x >> 5);
    int lane = threadIdx.x & 31;
    int stripsM = (M + 15) >> 4;
    if (strip >= stripsM) return;                   // wave-uniform: EXEC all-ones

    int m0 = strip << 4;
    int half = lane >> 4, lm = lane & 15;
    int rA = m0 + lm; if (rA > M - 1) rA = M - 1;   // clamp (keeps EXEC full)
    const float* arow = A + (size_t)rA * lda + 2 * half;

    v8f acc[TN];
#pragma unroll
    for (int t = 0; t < TN; ++t)
        acc[t] = (v8f){0.f, 0.f, 0.f, 0.f, 0.f, 0.f, 0.f, 0.f};

    v2f a_cur = *(const v2f*)arow;                  // k0 = 0 fragment
    int k0 = 0;
    for (; k0 + 4 < Kpad; k0 += 4) {
        v2f a_next = *(const v2f*)(arow + k0 + 4);  // pipeline next A fragment
        const v2f* brow = ldsB + ((k0 >> 1) + half) * Npad + lm;
#pragma unroll
        for (int t = 0; t < TN; ++t)
            acc[t] = __builtin_amdgcn_wmma_f32_16x16x4_f32(false, a_cur, false,
                                                           brow[t * 16], (short)0,
                                                           acc[t], false, false);
        a_cur = a_next;
    }
    {   // final K-step
        const v2f* brow = ldsB + ((k0 >> 1) + half) * Npad + lm;
#pragma unroll
        for (int t = 0; t < TN; ++t)
            acc[t] = __builtin_amdgcn_wmma_f32_16x16x4_f32(false, a_cur, false,
                                                           brow[t * 16], (short)0,
                                                           acc[t], false, false);
    }

#pragma unroll
    for (int t = 0; t < TN; ++t) {
        float* dcol = D + (size_t)m0 * ldd + t * 16 + lm;
#pragma unroll
        for (int v = 0; v < 8; ++v) {
            int row = v + 8 * half;
            if (m0 + row < M) dcol[(size_t)row * ldd] = acc[t][v];
        }
    }
}

// ---------------------------------------------------------------------------
// Edge scatter: one wave per edge; lane c handles channels c, c+32, ...
// agg[dst][c] += h[src][c] * dinv[src]*dinv[dst]
// ---------------------------------------------------------------------------
__global__ void gcn_scatter(const long long* __restrict__ src,
                            const long long* __restrict__ dst,
                            const float* __restrict__ dinv,
                            const float* __restrict__ h, int ldh, int C,
                            float* __restrict__ agg, long long E) {
    long long wave = ((long long)blockIdx.x * blockDim.x + threadIdx.x) >> 5;
    int lane = threadIdx.x & 31;
    long long nWaves = ((long long)gridDim.x * blockDim.x) >> 5;
    for (long long e = wave; e < E; e += nWaves) {
        int s = (int)src[e];
        int d = (int)dst[e];
        // prefetch next edge's source row (global_prefetch_b8)
        long long en = e + nWaves;
        if (en < E) {
            int sn = (int)src[en];
            __builtin_prefetch(h + (size_t)sn * ldh + lane, 0, 0);
        }
        float norm = dinv[s] * dinv[d];
        const float* hr = h + (size_t)s * ldh;
        float* ar = agg + (size_t)d * ldh;
        for (int ch = lane; ch < C; ch += 32)
            atomicAdd(&ar[ch], hr[ch] * norm);
    }
}

// ---------------------------------------------------------------------------
// out[n][c] = relu(agg[n][c] + h[n][c]*dinv[n]^2 + bias[c]), pad cols -> 0
// (in place: out == agg)
// ---------------------------------------------------------------------------
__global__ void gcn_selfloop_bias_relu(float* __restrict__ agg,
                                       const float* __restrict__ h,
                                       const float* __restrict__ dinv,
                                       const float* __restrict__ bias,
                                       int N, int ldh, int C) {
    size_t idx = (size_t)blockIdx.x * blockDim.x + threadIdx.x;
    size_t total = (size_t)N * ldh;
    size_t stride = (size_t)gridDim.x * blockDim.x;
    for (; idx < total; idx += stride) {
        int c = (int)(idx % ldh);
        int nd = (int)(idx / ldh);
        float v = 0.f;
        if (c < C) {
            float di = dinv[nd];
            v = fmaxf(agg[idx] + h[idx] * di * di + bias[c], 0.f);
        }
        agg[idx] = v;
    }
}

// ---------------------------------------------------------------------------
// Final: v = agg + h*dinv^2 + b2, then log_softmax over C=40 channels.
// One wave per node; wave32 shfl_xor reductions.
// ---------------------------------------------------------------------------
__global__ void gcn_final_logsoftmax(const float* __restrict__ agg,
                                     const float* __restrict__ h,
                                     const float* __restrict__ dinv,
                                     const float* __restrict__ bias,
                                     float* __restrict__ out,
                                     int N, int ldh, int C) {
    int wave = (blockIdx.x * blockDim.x + threadIdx.x) >> 5;
    int lane = threadIdx.x & 31;
    int nWaves = (gridDim.x * blockDim.x) >> 5;
    for (int nd = wave; nd < N; nd += nWaves) {
        float di = dinv[nd];
        float dsq = di * di;
        const float* ar = agg + (size_t)nd * ldh;
        const float* hr = h + (size_t)nd * ldh;
        float v0 = ar[lane] + hr[lane] * dsq + bias[lane];   // lane < 32 < C=40
        int c1 = lane + 32;
        bool has1 = (c1 < C);
        float v1 = has1 ? (ar[c1] + hr[c1] * dsq + bias[c1]) : -INFINITY;
        float m = fmaxf(v0, v1);
        for (int off = 16; off; off >>= 1)
            m = fmaxf(m, __shfl_xor(m, off, 32));
        float s = expf(v0 - m) + (has1 ? expf(v1 - m) : 0.f);
        for (int off = 16; off; off >>= 1)
            s += __shfl_xor(s, off, 32);
        float ls = m + logf(s);
        float* orow = out + (size_t)nd * C;
        orow[lane] = v0 - ls;
        if (has1) orow[c1] = v1 - ls;
    }
}

// ---------------------------------------------------------------------------
extern "C" void kernel_launch(void* const* d_in, const int* in_sizes, int n_in,
                              void* d_out, int out_size, void* d_ws, size_t ws_size,
                              hipStream_t stream) {
    const int IN_CH = 128, HID = 75, HIDP = 80, OUT = 40, OUTP = 48;

    const float* x = (const float*)d_in[0];
    const long long* ei = (const long long*)d_in[1];
    const float* W1 = (const float*)d_in[2];
    const float* b1 = (const float*)d_in[3];
    const float* W2 = (const float*)d_in[4];
    const float* b2 = (const float*)d_in[5];

    int N = in_sizes[0] / IN_CH;                 // 100000
    long long E = (long long)in_sizes[1] / 2;    // 3200000
    const long long* src = ei;
    const long long* dstv = ei + E;

    // workspace layout
    char* w = (char*)d_ws;
    size_t oH = ((size_t)N * 4 + 255) & ~(size_t)255;        // after dinv
    size_t oAgg = oH + (size_t)N * HIDP * 4;                 // after h buffer
    float* dinv = (float*)w;
    float* hbuf = (float*)(w + oH);     // h1 [N x 80], later h2 [N x 48]
    float* agg  = (float*)(w + oAgg);   // agg1/a1 [N x 80], later agg2 [N x 48]
    (void)ws_size; (void)n_in; (void)out_size;

    // ---- degree / dinv ----
    hipMemsetAsync(dinv, 0, (size_t)N * 4, stream);
    gcn_degree<<<4096, 256, 0, stream>>>(dstv, dinv, E);
    gcn_dinv<<<(N + 255) / 256, 256, 0, stream>>>(dinv, N);

    int strips = (N + 15) / 16;                   // 6250
    int gemmBlocks = (strips + 7) / 8;            // 8 waves (strips) per block

    // ---- layer 1: h1 = x @ W1 (WMMA f32, B in LDS, 5 N-tiles per wave) ----
    gcn_wmma_gemm<5><<<gemmBlocks, 256, (size_t)IN_CH * HIDP * 4, stream>>>(
        x, IN_CH, W1, HID, HID, IN_CH, IN_CH, hbuf, HIDP, N);

    hipMemsetAsync(agg, 0, (size_t)N * HIDP * 4, stream);
    gcn_scatter<<<4096, 256, 0, stream>>>(src, dstv, dinv, hbuf, HIDP, HID, agg, E);
    gcn_selfloop_bias_relu<<<4096, 256, 0, stream>>>(agg, hbuf, dinv, b1, N, HIDP, HID);

    // ---- layer 2: h2 = a1 @ W2 (WMMA f32, K padded 75->80, 3 N-tiles) ----
    gcn_wmma_gemm<3><<<gemmBlocks, 256, (size_t)HIDP * OUTP * 4, stream>>>(
        agg, HIDP, W2, OUT, OUT, HID, HIDP, hbuf, OUTP, N);

    hipMemsetAsync(agg, 0, (size_t)N * OUTP * 4, stream);
    gcn_scatter<<<4096, 256, 0, stream>>>(src, dstv, dinv, hbuf, OUTP, OUT, agg, E);

    // ---- self-loop + bias + log_softmax -> d_out ----
    gcn_final_logsoftmax<<<(N + 7) / 8, 256, 0, stream>>>(agg, hbuf, dinv, b2,
                                                          (float*)d_out, N, OUTP, OUT);
}